// weights_13211319402848
// MI455X (gfx1250) — compile-verified
//
#include <hip/hip_runtime.h>
#include <math.h>

// ---------------------------------------------------------------------------
// Problem constants (from reference setup_inputs)
// ---------------------------------------------------------------------------
constexpr int  B_ = 8, T_ = 200, N_ = 128, H_ = 72, W_ = 128;
constexpr long long BT_   = (long long)B_ * T_;        // 1600
constexpr long long NSQ_  = (long long)N_ * N_;        // 16384
constexpr long long TOTAL = BT_ * NSQ_;                // 26,214,400 elements

// Workspace layout (bytes)
constexpr size_t U_OFF     = 0;                               // float4 per (bt,n)
constexpr size_t U_BYTES   = (size_t)BT_ * N_ * 16;           // 3,276,800
constexpr size_t NRM_OFF   = U_OFF + U_BYTES;                 // float per (bt,n)
constexpr size_t NRM_BYTES = (size_t)BT_ * N_ * 4;            //   819,200
constexpr size_t S_OFF     = NRM_OFF + NRM_BYTES;             // 6 x u32 scalars
constexpr size_t DIST_OFF  = ((S_OFF + 6 * 4 + 255) / 256) * 256; // 104.9 MB dist

typedef __attribute__((ext_vector_type(2))) float v2f;
typedef __attribute__((ext_vector_type(8))) float v8f;

// Order-preserving float <-> uint encoding so we can use atomicMin/Max(u32)
__device__ __forceinline__ unsigned encf(float f) {
  unsigned u = __float_as_uint(f);
  return (u & 0x80000000u) ? ~u : (u | 0x80000000u);
}
__device__ __forceinline__ float decf(unsigned k) {
  return (k & 0x80000000u) ? __uint_as_float(k & 0x7fffffffu)
                           : __uint_as_float(~k);
}

// Block-wide min/max reduction, then one encoded atomic per block.
__device__ __forceinline__ void block_minmax_commit(float vmin, float vmax,
                                                    unsigned* gmin,
                                                    unsigned* gmax) {
  __shared__ float smin[256];
  __shared__ float smax[256];
  const int t = threadIdx.x;
  smin[t] = vmin;
  smax[t] = vmax;
  __syncthreads();
  for (int s = 128; s > 0; s >>= 1) {
    if (t < s) {
      smin[t] = fminf(smin[t], smin[t + s]);
      smax[t] = fmaxf(smax[t], smax[t + s]);
    }
    __syncthreads();
  }
  if (t == 0) {
    atomicMin(gmin, encf(smin[0]));
    atomicMax(gmax, encf(smax[0]));
  }
}

// ---------------------------------------------------------------------------
// K0: init the 6 reduction scalars (min slots -> 0xFFFFFFFF, max slots -> 0)
// ---------------------------------------------------------------------------
__global__ void k0_init(unsigned* __restrict__ s) {
  if (threadIdx.x < 6) s[threadIdx.x] = (threadIdx.x & 1) ? 0u : 0xFFFFFFFFu;
}

// ---------------------------------------------------------------------------
// K1: stable compaction (drop class==2, pack front, zero-fill), centers,
//     depth gather.  u = (cx/1450, cy/1450, depth[yi][xi]/255, 0),
//     nrm = |u|^2.  Zero-filled rows index depth[0][0] (matches reference).
//     One thread per (b,t).
// ---------------------------------------------------------------------------
__global__ void k1_compact(const float* __restrict__ det_g,
                           const float* __restrict__ depth_g,
                           float4* __restrict__ U, float* __restrict__ NRM) {
  const long long bt = (long long)blockIdx.x * blockDim.x + threadIdx.x;
  if (bt >= BT_) return;
  const float* det = det_g + (size_t)bt * N_ * 6;
  const float* dep = depth_g + (size_t)bt * H_ * W_;
  float4* u = U + (size_t)bt * N_;
  float* nr = NRM + (size_t)bt * N_;
  int k = 0;
  for (int r = 0; r < N_; ++r) {
    const float* row = det + r * 6;
    if (row[4] != 2.0f) {
      const float cx = (row[0] + row[2]) * 0.5f;
      const float cy = (row[1] + row[3]) * 0.5f;
      const int xi = (int)(cx / 10.0f);
      const int yi = (int)(cy / 10.0f);
      const float dv = dep[yi * W_ + xi] * (1.0f / 255.0f);
      const float ux = cx * (1.0f / 1450.0f);
      const float uy = cy * (1.0f / 1450.0f);
      float4 v;
      v.x = ux; v.y = uy; v.z = dv; v.w = 0.0f;   // w==0 keeps K=4 WMMA exact
      u[k] = v;
      nr[k] = ux * ux + uy * uy + dv * dv;
      ++k;
    }
  }
  const float dv0 = dep[0] * (1.0f / 255.0f);     // zeroed rows -> depth[0][0]
  for (; k < N_; ++k) {
    float4 v;
    v.x = 0.0f; v.y = 0.0f; v.z = dv0; v.w = 0.0f;
    u[k] = v;
    nr[k] = dv0 * dv0;
  }
}

// ---------------------------------------------------------------------------
// K2: dist[b,t,i,j] = sqrt(max(0, |u_i|^2 + |u_j|^2 - 2 u_i.u_j)) via
//     V_WMMA_F32_16X16X4_F32 Gram tiles.  One wave per 16x16 tile,
//     8 waves/block, 8 blocks per (b,t).  Also global min/max of dist.
// ---------------------------------------------------------------------------
__global__ void __launch_bounds__(256) k2_dist(const float4* __restrict__ U,
                                               const float* __restrict__ NRM,
                                               float* __restrict__ DIST,
                                               unsigned* __restrict__ s) {
  const int bt   = blockIdx.x >> 3;
  const int wave = threadIdx.x >> 5;
  const int lane = threadIdx.x & 31;
  const int tile = ((blockIdx.x & 7) << 3) | wave;  // 0..63
  const int ti = tile >> 3, tj = tile & 7;
  const int hl = lane >> 4;                         // K half: 0 -> K=0,1 ; 1 -> K=2,3
  const int m = ti * 16 + (lane & 15);              // A row for this lane
  const int n = tj * 16 + (lane & 15);              // B col for this lane

  // A: 16x4 f32, lane L holds (M = L&15, K = 2*hl .. 2*hl+1).  B symmetric
  // (Gram: B[k][n] = u_n[k]).  u.w == 0 so K=4 adds nothing.
  const float4 um = U[(size_t)bt * N_ + m];
  const float4 un = U[(size_t)bt * N_ + n];
  v2f a, b;
  a.x = hl ? um.z : um.x;  a.y = hl ? um.w : um.y;
  b.x = hl ? un.z : un.x;  b.y = hl ? un.w : un.y;

  v8f c = {};
  c = __builtin_amdgcn_wmma_f32_16x16x4_f32(false, a, false, b, (short)0, c,
                                            false, false);

  // C/D layout: VGPR v -> row = 16*ti + v + 8*hl, col = 16*tj + (lane&15)
  const float ncol   = NRM[(size_t)bt * N_ + n];
  const int rowBase  = ti * 16 + (hl << 3);
  float vmin = 3.4e38f, vmax = -3.4e38f;
#pragma unroll
  for (int v = 0; v < 8; ++v) {
    const int row  = rowBase + v;
    const float d2 = NRM[(size_t)bt * N_ + row] + ncol - 2.0f * c[v];
    const float d  = d2 > 0.0f ? sqrtf(d2) : 0.0f;   // safe sqrt
    DIST[(size_t)bt * NSQ_ + (size_t)row * N_ + n] = d;
    vmin = fminf(vmin, d);
    vmax = fmaxf(vmax, d);
  }
  block_minmax_commit(vmin, vmax, s + 0, s + 1);
}

// ---------------------------------------------------------------------------
// K3: global min/max over POSITIVE velocities (vel = dist[t]-dist[t-1], 0 @ t=0)
// ---------------------------------------------------------------------------
__global__ void k3_vel(const float* __restrict__ DIST,
                       unsigned* __restrict__ s) {
  float vmin = 3.4e38f, vmax = -3.4e38f;
  const long long stride = (long long)gridDim.x * blockDim.x;
  for (long long idx = (long long)blockIdx.x * blockDim.x + threadIdx.x;
       idx < TOTAL; idx += stride) {
    const long long bt = idx / NSQ_;
    const int t = (int)(bt % T_);
    if (t == 0) continue;                       // vel==0, never positive
    const float vel = DIST[idx] - DIST[idx - NSQ_];
    if (vel > 0.0f) {
      vmin = fminf(vmin, vel);
      vmax = fmaxf(vmax, vel);
    }
  }
  block_minmax_commit(vmin, vmax, s + 2, s + 3);
}

// ---------------------------------------------------------------------------
// K4: w = 0.8*exp(-normalize(dist)) + 0.2*vel_norm(vel); write w to out,
//     reduce global min/max of w.
// ---------------------------------------------------------------------------
__global__ void k4_w(const float* __restrict__ DIST, float* __restrict__ out,
                     const unsigned* __restrict__ s, unsigned* __restrict__ sw) {
  const float dmn  = decf(s[0]);
  const float drng = decf(s[1]) - dmn;
  const float pmn  = decf(s[2]);
  const float prng = decf(s[3]) - pmn;
  float vmin = 3.4e38f, vmax = -3.4e38f;
  const long long stride = (long long)gridDim.x * blockDim.x;
  for (long long idx = (long long)blockIdx.x * blockDim.x + threadIdx.x;
       idx < TOTAL; idx += stride) {
    const long long bt = idx / NSQ_;
    const int t = (int)(bt % T_);
    const float d  = DIST[idx];
    const float dn = (drng != 0.0f) ? (d - dmn) / drng : 0.0f;
    const float vel = (t == 0) ? 0.0f : d - DIST[idx - NSQ_];
    const float scl = (prng != 0.0f) ? (vel - pmn) / prng : 1.0f;
    const float vn  = (vel > 0.0f) ? scl : 0.0f;
    const float w   = 0.8f * expf(-dn) + 0.2f * vn;
    out[idx] = w;
    vmin = fminf(vmin, w);
    vmax = fmaxf(vmax, w);
  }
  block_minmax_commit(vmin, vmax, sw + 0, sw + 1);
}

// ---------------------------------------------------------------------------
// K5: final in-place normalize of out
// ---------------------------------------------------------------------------
__global__ void k5_norm(float* __restrict__ out,
                        const unsigned* __restrict__ sw) {
  const float mn  = decf(sw[0]);
  const float rng = decf(sw[1]) - mn;
  const long long stride = (long long)gridDim.x * blockDim.x;
  for (long long idx = (long long)blockIdx.x * blockDim.x + threadIdx.x;
       idx < TOTAL; idx += stride) {
    const float w = out[idx];
    out[idx] = (rng != 0.0f) ? (w - mn) / rng : 0.0f;
  }
}

// ---------------------------------------------------------------------------
extern "C" void kernel_launch(void* const* d_in, const int* in_sizes, int n_in,
                              void* d_out, int out_size, void* d_ws,
                              size_t ws_size, hipStream_t stream) {
  const float* det   = (const float*)d_in[0];   // detection [8,200,128,6]
  const float* depth = (const float*)d_in[1];   // depth     [8,200,72,128]
  float* out = (float*)d_out;                   // w         [8,200,128,128]

  char* ws = (char*)d_ws;
  float4*   U    = (float4*)(ws + U_OFF);
  float*    NRM  = (float*)(ws + NRM_OFF);
  unsigned* S    = (unsigned*)(ws + S_OFF);     // [0..5] encoded min/max
  float*    DIST = (float*)(ws + DIST_OFF);     // 104.9 MB

  k0_init<<<1, 32, 0, stream>>>(S);
  k1_compact<<<(int)((BT_ + 255) / 256), 256, 0, stream>>>(det, depth, U, NRM);
  k2_dist<<<(int)(BT_ * 8), 256, 0, stream>>>(U, NRM, DIST, S);
  k3_vel<<<1024, 256, 0, stream>>>(DIST, S);
  k4_w<<<1024, 256, 0, stream>>>(DIST, out, S, S + 4);
  k5_norm<<<1024, 256, 0, stream>>>(out, S + 4);
}